// GAT_MaxMargin_68281390072286
// MI455X (gfx1250) — compile-verified
//
#include <hip/hip_runtime.h>
#include <cstdint>
#include <cstddef>

#define B_ 2048
#define V_ 64
#define D_ 768
#define H_ 256
#define R_ 16

typedef float v2f __attribute__((ext_vector_type(2)));
typedef float v8f __attribute__((ext_vector_type(8)));

// ---------------------------------------------------------------------------
// Kernel 1: per-relation precompute (r = 0..15).
//   u[r,d]   = sum_h Wv[r,h,d] * a1[r,h]            (a1 = Aw[r, :H])
//   rel_t[h] = sum_d Wv[r,h,d] * prop[r,d] + bv[r,h]
//   c[r]     = sum_h bv[r,h]*a1[h] + sum_h rel_t[h]*a2[h] + ab[r]
// ---------------------------------------------------------------------------
__global__ __launch_bounds__(256) void precompute_k(
    const float* __restrict__ prop, const float* __restrict__ Wv,
    const float* __restrict__ bv,   const float* __restrict__ Aw,
    const float* __restrict__ ab,
    float* __restrict__ u_ws, float* __restrict__ c_ws)
{
    const int r = blockIdx.x;
    const int h = threadIdx.x;            // 0..255 == H_
    __shared__ float a1[H_];
    __shared__ float red[H_];

    const float a1h = Aw[r * 2 * H_ + h];
    const float a2h = Aw[r * 2 * H_ + H_ + h];
    a1[h] = a1h;

    const float* wrow = Wv + ((size_t)r * H_ + h) * D_;
    const float* pr   = prop + (size_t)r * D_;
    float acc = 0.f;
    for (int d = 0; d < D_; ++d) acc = fmaf(wrow[d], pr[d], acc);
    const float bvh = bv[r * H_ + h];
    acc += bvh;                           // rel_t[r,h]
    red[h] = bvh * a1h + acc * a2h;
    __syncthreads();
    for (int s = H_ / 2; s > 0; s >>= 1) {
        if (h < s) red[h] += red[h + s];
        __syncthreads();
    }
    if (h == 0) c_ws[r] = red[0] + ab[r];

    // u[r, d] — coalesced over d across threads
    for (int d = h; d < D_; d += H_) {
        float s = 0.f;
        for (int hh = 0; hh < H_; ++hh)
            s = fmaf(Wv[((size_t)r * H_ + hh) * D_ + d], a1[hh], s);
        u_ws[r * D_ + d] = s;
    }
}

// ---------------------------------------------------------------------------
// Kernel 2: main streaming kernel. One block per batch element b.
// 8 waves: wave = {tensor(head/tail) x 4 v-tiles of 16 rows}.
// Each wave accumulates D[m,n] over K=768 in steps of 4 with
// V_WMMA_F32_16X16X4_F32. B panel in LDS is class-indexed so the inner
// loop is branch-free: class 0 = u, 1 = Ow[:D], 2 = Ow[D:], 3 = zeros.
// ---------------------------------------------------------------------------
__global__ __launch_bounds__(256) void gat_main_k(
    const float* __restrict__ head, const float* __restrict__ tail,
    const float* __restrict__ Ow,   const float* __restrict__ ob,
    const int*  __restrict__ rc_ids,
    const float* __restrict__ u_ws, const float* __restrict__ c_ws,
    float* __restrict__ scores)
{
    __shared__ float bpanel[4 * D_];   // [u | w1 | w2 | zeros]
    __shared__ float pq[4][V_];        // [p_head, q_head, p_tail, q_tail]

    const int b   = blockIdx.x;
    const int tid = threadIdx.x;
    const int r   = rc_ids[b];

    const float* owr = Ow + (size_t)r * 2 * D_;
    for (int d = tid; d < D_; d += 256) {
        bpanel[d]          = u_ws[r * D_ + d];
        bpanel[D_ + d]     = owr[d];
        bpanel[2 * D_ + d] = owr[D_ + d];
        bpanel[3 * D_ + d] = 0.f;
    }
    __syncthreads();

    const int wave   = tid >> 5;
    const int lane   = tid & 31;
    const int tensor = wave >> 2;       // 0 = head, 1 = tail
    const int tile   = wave & 3;        // v rows [tile*16, tile*16+16)
    const int mrow   = lane & 15;       // A-matrix M index == B-matrix N index
    const int half   = lane >> 4;       // 0: K=k0,k0+1 ; 1: K=k0+2,k0+3

    const float* vals = tensor ? tail : head;
    // A-matrix 16x4 f32 layout: lane m holds {K0,K1}; lane 16+m holds {K2,K3}
    const float* rowp = vals + ((size_t)b * V_ + tile * 16 + mrow) * D_ + half * 2;

    // Per-lane B class: N==0 -> u, N==1 -> Ow half (per tensor), else zeros.
    const int ncol = mrow;
    const int cls  = (ncol == 0) ? 0 : ((ncol == 1) ? (tensor ? 2 : 1) : 3);
    const float* bsrc = bpanel + cls * D_ + half * 2;

    v8f acc0 = {0.f, 0.f, 0.f, 0.f, 0.f, 0.f, 0.f, 0.f};
    v8f acc1 = {0.f, 0.f, 0.f, 0.f, 0.f, 0.f, 0.f, 0.f};

    for (int k0 = 0; k0 < D_; k0 += 64) {
        __builtin_prefetch(rowp + k0 + 256, 0, 3);
        #pragma unroll
        for (int kk = 0; kk < 64; kk += 8) {
            v2f a0 = *(const v2f*)(rowp + k0 + kk);
            v2f b0 = *(const v2f*)(bsrc + k0 + kk);
            v2f a1 = *(const v2f*)(rowp + k0 + kk + 4);
            v2f b1 = *(const v2f*)(bsrc + k0 + kk + 4);
            acc0 = __builtin_amdgcn_wmma_f32_16x16x4_f32(
                false, a0, false, b0, (short)0, acc0, false, false);
            acc1 = __builtin_amdgcn_wmma_f32_16x16x4_f32(
                false, a1, false, b1, (short)0, acc1, false, false);
        }
    }
    v8f acc = acc0 + acc1;

    // C/D layout: VGPR j -> M=j (lanes 0-15, N=lane), M=8+j (lanes 16-31, N=lane-16)
    if (ncol < 2) {
        #pragma unroll
        for (int j = 0; j < 8; ++j)
            pq[tensor * 2 + ncol][tile * 16 + half * 8 + j] = acc[j];
    }
    __syncthreads();

    if (tid == 0) {
        const float c  = c_ws[r];
        float sc = ob[r];
        for (int t = 0; t < 2; ++t) {
            float emax = -3.0e38f;
            for (int v = 0; v < V_; ++v) {
                float e = pq[t * 2][v] + c;
                e = e > 0.f ? e : 0.01f * e;          // leaky_relu
                emax = fmaxf(emax, e);
            }
            float den = 0.f, num = 0.f;
            for (int v = 0; v < V_; ++v) {
                float e = pq[t * 2][v] + c;
                e = e > 0.f ? e : 0.01f * e;
                float x = __expf(e - emax);
                den += x;
                num = fmaf(x, pq[t * 2 + 1][v], num);
            }
            sc += num / den;
        }
        scores[b] = sc;
    }
}

// ---------------------------------------------------------------------------
// Kernel 3: margin loss over pairs. pair i: neg=scores[2i], pos=scores[2i+1]
// ---------------------------------------------------------------------------
__global__ __launch_bounds__(256) void loss_k(float* __restrict__ out)
{
    __shared__ float red[256];
    const int tid = threadIdx.x;
    float s = 0.f;
    for (int i = tid; i < B_ / 2; i += 256) {
        float diff = out[2 * i + 1] - out[2 * i];
        float v = 1.0f - diff;
        s += v > 0.f ? v : 0.f;
    }
    red[tid] = s;
    __syncthreads();
    for (int st = 128; st > 0; st >>= 1) {
        if (tid < st) red[tid] += red[tid + st];
        __syncthreads();
    }
    if (tid == 0) out[B_] = red[0] / (float)(B_ / 2);
}

extern "C" void kernel_launch(void* const* d_in, const int* in_sizes, int n_in,
                              void* d_out, int out_size, void* d_ws, size_t ws_size,
                              hipStream_t stream) {
    const float* head = (const float*)d_in[0];
    const float* tail = (const float*)d_in[1];
    const float* prop = (const float*)d_in[2];
    const float* Wv   = (const float*)d_in[3];
    const float* bv   = (const float*)d_in[4];
    const float* Aw   = (const float*)d_in[5];
    const float* ab   = (const float*)d_in[6];
    const float* Ow   = (const float*)d_in[7];
    const float* ob   = (const float*)d_in[8];
    const int*   rc   = (const int*)d_in[9];
    (void)in_sizes; (void)n_in; (void)out_size; (void)ws_size;

    float* u_ws = (float*)d_ws;            // 16*768 floats
    float* c_ws = u_ws + R_ * D_;          // 16 floats
    float* out  = (float*)d_out;           // scores[2048] + loss[1]

    precompute_k<<<R_, 256, 0, stream>>>(prop, Wv, bv, Aw, ab, u_ws, c_ws);
    gat_main_k<<<B_, 256, 0, stream>>>(head, tail, Ow, ob, rc, u_ws, c_ws, out);
    loss_k<<<1, 256, 0, stream>>>(out);
}